// OhemCE_8194797600923
// MI455X (gfx1250) — compile-verified
//
#include <hip/hip_runtime.h>

// Problem constants (match the reference).
#define BB 8
#define CC 19
#define HH 512
#define WW 1024
#define HW (HH * WW)                 // 524288 pixels per (b,c) plane
#define NPIX ((size_t)BB * HW)       // 4194304 pixels
#define NGRP (NPIX / 4)              // 1048576 float4 pixel-groups
#define GPB (HW / 4)                 // 131072 groups per batch image
#define NBLK 2048
#define NTHR 256

typedef float f32x4 __attribute__((ext_vector_type(4)));
typedef int   i32x4 __attribute__((ext_vector_type(4)));
typedef float v2f   __attribute__((ext_vector_type(2)));
typedef float v8f   __attribute__((ext_vector_type(8)));

// Sum x across all 32 lanes of a wave using V_WMMA_F32_16X16X4_F32.
// A = ones(16x4), B row0/row2 = lane values, rows 1/3 = 0:
//   D[m][n] = x[n] + x[n+16]  for every row m, so D's first accumulator VGPR
//   holds colsum[L%16] in every lane L.  4 xor-shuffles finish the 16-way sum.
// Caller must guarantee the wave's EXEC is all 1s (uniform control flow).
__device__ __forceinline__ float wave_sum_wmma(float x) {
  v2f a; a.x = 1.0f; a.y = 1.0f;   // A-matrix: all ones
  v2f b; b.x = x;    b.y = 0.0f;   // B-matrix: lane partials in K=0/K=2 rows
  v8f c = {};
  c = __builtin_amdgcn_wmma_f32_16x16x4_f32(
      /*neg_a=*/false, a, /*neg_b=*/false, b,
      /*c_mod=*/(short)0, c, /*reuse_a=*/false, /*reuse_b=*/false);
  float s = c[0];                  // colsum[lane % 16]
  s += __shfl_xor(s, 1, 32);
  s += __shfl_xor(s, 2, 32);
  s += __shfl_xor(s, 4, 32);
  s += __shfl_xor(s, 8, 32);
  return s;                        // full 32-lane sum, in every lane
}

// Stage 1: each thread streams float4 pixel-groups (19 NT B128 loads + 1 label
// load per group), computes masked NLL for 4 pixels, accumulates, then the
// block reduces to one partial in d_ws.
__global__ void __launch_bounds__(NTHR)
ce_partial_kernel(const float* __restrict__ logits,
                  const int*   __restrict__ labels,
                  float*       __restrict__ partial) {
  const size_t tid    = (size_t)blockIdx.x * blockDim.x + threadIdx.x;
  const size_t stride = (size_t)gridDim.x * blockDim.x;

  float acc = 0.0f;

  for (size_t g = tid; g < NGRP; g += stride) {
    const size_t bimg = g / GPB;
    const size_t s4   = g - bimg * GPB;
    const f32x4* lg   = (const f32x4*)logits + bimg * (size_t)CC * (HW / 4) + s4;

    // Prefetch next grid-stride iteration's class-0 line (global_prefetch_b8).
    const size_t gn = g + stride;
    if (gn < NGRP) {
      const size_t bn = gn / GPB, sn = gn - bn * GPB;
      __builtin_prefetch((const void*)((const f32x4*)logits +
                                       bn * (size_t)CC * (HW / 4) + sn), 0, 0);
    }

    // Read all 19 class values for 4 pixels exactly once, non-temporally.
    f32x4 v[CC];
#pragma unroll
    for (int c = 0; c < CC; ++c)
      v[c] = __builtin_nontemporal_load(lg + (size_t)c * (HW / 4));

    const i32x4 lab = __builtin_nontemporal_load(
        (const i32x4*)labels + bimg * (HW / 4) + s4);

#pragma unroll
    for (int j = 0; j < 4; ++j) {
      const int l    = lab[j];
      const int safe = l < 0 ? 0 : (l > (CC - 1) ? (CC - 1) : l);

      float m = v[0][j];
#pragma unroll
      for (int c = 1; c < CC; ++c) m = fmaxf(m, v[c][j]);

      float se = 0.0f, xl = 0.0f;
#pragma unroll
      for (int c = 0; c < CC; ++c) {
        se += __expf(v[c][j] - m);
        if (c == safe) xl = v[c][j];   // compile-time c vs runtime safe -> cndmask
      }
      const float nll = __logf(se) + m - xl;
      acc += (l != -1) ? nll : 0.0f;
    }
  }

  // Block reduction: per-wave WMMA fold, then first wave folds the 8 wave sums.
  __shared__ float smem[NTHR / 32];
  const int wave = threadIdx.x >> 5;
  const int lane = threadIdx.x & 31;

  const float ws = wave_sum_wmma(acc);   // full block => EXEC all 1s
  if (lane == 0) smem[wave] = ws;
  __syncthreads();

  if (wave == 0) {                       // wave-uniform branch: EXEC all 1s
    float t = (lane < (NTHR / 32)) ? smem[lane] : 0.0f;
    t = wave_sum_wmma(t);
    if (lane == 0) partial[blockIdx.x] = t;
  }
}

// Stage 2: one block deterministically reduces the 2048 partials and writes
// mean = sum / NPIX.
__global__ void __launch_bounds__(NTHR)
ce_final_kernel(const float* __restrict__ partial,
                float*       __restrict__ out,
                int n) {
  float acc = 0.0f;
  for (int i = threadIdx.x; i < n; i += NTHR) acc += partial[i];

  __shared__ float smem[NTHR / 32];
  const int wave = threadIdx.x >> 5;
  const int lane = threadIdx.x & 31;

  const float ws = wave_sum_wmma(acc);
  if (lane == 0) smem[wave] = ws;
  __syncthreads();

  if (wave == 0) {
    float t = (lane < (NTHR / 32)) ? smem[lane] : 0.0f;
    t = wave_sum_wmma(t);
    if (lane == 0) out[0] = t * (1.0f / (float)NPIX);
  }
}

extern "C" void kernel_launch(void* const* d_in, const int* in_sizes, int n_in,
                              void* d_out, int out_size, void* d_ws, size_t ws_size,
                              hipStream_t stream) {
  (void)in_sizes; (void)n_in; (void)out_size; (void)ws_size;
  const float* logits = (const float*)d_in[0];
  const int*   labels = (const int*)d_in[1];
  // d_in[2] (smooth_labels) is dead code in the reference forward: unused.
  float* partial = (float*)d_ws;         // needs NBLK*4 = 8 KB of scratch

  ce_partial_kernel<<<NBLK, NTHR, 0, stream>>>(logits, labels, partial);
  ce_final_kernel<<<1, NTHR, 0, stream>>>(partial, (float*)d_out, NBLK);
}